// Cen_IoU_Loss_74440373175045
// MI455X (gfx1250) — compile-verified
//
#include <hip/hip_runtime.h>
#include <hip/hip_bf16.h>

// Pipeline:
//  K0  build sortable keys from IoU (descending) + payload = centerness bits
//  4 x (histogram -> global digit scan -> stable scatter)   [LSD radix, 8-bit digits]
//  K1  per-block sums of e = exp(-s) over sorted order (double)
//  K2  exclusive scan of block sums + grand total T (double)
//  K3  per-element suffix = T - inclusive_prefix, accumulate loss terms (WMMA block scan)
//  K4  final reduction, divide by (n-1)
//
// CDNA5 paths: v_wmma_f32_16x16x4_f32 block scans (radix ranking + prefix stage),
// global_load_async_to_lds_b128 tile staging, global_store_async_from_lds_b32 scatter.
//
// Workspace layout (n = 4,194,304): 4 x n u32 ping-pong (64MB) + hist (1MB) + ~24KB doubles.

typedef float v2f __attribute__((ext_vector_type(2)));
typedef float v8f __attribute__((ext_vector_type(8)));
typedef int   v4i __attribute__((vector_size(4 * sizeof(int))));

#define RDX_TILE 4096
#define RDX_THREADS 256
#define RDX_ITEMS 16

#if defined(__has_builtin)
#  if __has_builtin(__builtin_amdgcn_global_load_async_to_lds_b128) && \
      __has_builtin(__builtin_amdgcn_s_wait_asynccnt)
#    define CIOU_ASYNC_LOAD 1
#  endif
#  if __has_builtin(__builtin_amdgcn_global_store_async_from_lds_b32) && \
      __has_builtin(__builtin_amdgcn_s_wait_asynccnt)
#    define CIOU_ASYNC_STORE 1
#  endif
#endif

// Exactly-typed pointer casts for the async builtins:
//   arg0: global side  (AS(1), printed as __device__ in HIP diagnostics)
//   arg1: LDS side     (AS(3))
typedef __attribute__((address_space(1))) v4i* glb_v4i_ptr;
typedef __attribute__((address_space(3))) v4i* lds_v4i_ptr;
typedef __attribute__((address_space(1))) int* glb_i32_ptr;
typedef __attribute__((address_space(3))) int* lds_i32_ptr;

// ---------------------------------------------------------------------------
// Block-wide inclusive scan of 256 floats using V_WMMA_F32_16X16X4_F32.
// X is laid out column-major: element t lives at (row = t%16, col = t/16).
// D = L(16x16, lower-triangular ones) * X, accumulated over four K=4 slices.
// Column fix-up (16 adds) done by one lane. Exact for integer-valued inputs.
// ---------------------------------------------------------------------------
__device__ static float blockScan256(float v, float* lbuf, float* lscan,
                                     float* colOff, float* totalOut) {
  const int t = threadIdx.x;
  lbuf[t] = v;
  __syncthreads();
#if defined(__has_builtin) && __has_builtin(__builtin_amdgcn_wmma_f32_16x16x4_f32)
  if (t < 32) {
    const int lane = t;
    const int m  = lane & 15;          // A row / B column (n)
    const int kb = (lane >> 4) << 1;   // K sub-index base: 0 (lanes 0-15) or 2 (lanes 16-31)
    v8f d = {};
#pragma unroll
    for (int c = 0; c < 4; ++c) {
      v2f a, bm;
      a[0]  = ((4 * c + kb + 0) <= m) ? 1.0f : 0.0f;   // L[:, 4c+kb]
      a[1]  = ((4 * c + kb + 1) <= m) ? 1.0f : 0.0f;   // L[:, 4c+kb+1]
      bm[0] = lbuf[m * 16 + 4 * c + kb + 0];           // X[4c+kb+0][n]
      bm[1] = lbuf[m * 16 + 4 * c + kb + 1];           // X[4c+kb+1][n]
      d = __builtin_amdgcn_wmma_f32_16x16x4_f32(false, a, false, bm,
                                                (short)0, d, false, false);
    }
#pragma unroll
    for (int j = 0; j < 8; ++j) {
      // D vgpr j: lanes 0-15 -> (M=j, N=lane); lanes 16-31 -> (M=j+8, N=lane-16)
      const int idx = (lane < 16) ? (16 * lane + j) : (16 * (lane - 16) + j + 8);
      lscan[idx] = d[j];
    }
  }
  __syncthreads();
  if (t == 0) {
    float run = 0.0f;
    for (int c = 0; c < 16; ++c) { colOff[c] = run; run += lscan[c * 16 + 15]; }
    colOff[16] = run;
  }
  __syncthreads();
  const float incl = lscan[t] + colOff[t >> 4];
  *totalOut = colOff[16];
  return incl;
#else
  if (t == 0) {
    float run = 0.0f;
    for (int i = 0; i < 256; ++i) { run += lbuf[i]; lscan[i] = run; }
    colOff[16] = run;
  }
  __syncthreads();
  *totalOut = colOff[16];
  return lscan[t];
#endif
}

// ---------------------------------------------------------------------------
// K0: IoU -> descending-order sortable u32 key, payload = centerness bits
// ---------------------------------------------------------------------------
__global__ void ciou_build_keys(const float* __restrict__ cen,
                                const float4* __restrict__ boxA,
                                const float4* __restrict__ boxB,
                                unsigned int* __restrict__ keys,
                                unsigned int* __restrict__ vals, int n) {
  const int i = blockIdx.x * blockDim.x + threadIdx.x;
  if (i >= n) return;
  const float4 p = boxA[i];   // (l, t, r, b)
  const float4 q = boxB[i];
  const float areaP = (p.x + p.z) * (p.y + p.w);
  const float areaQ = (q.x + q.z) * (q.y + q.w);
  const float wInt  = fminf(p.x, q.x) + fminf(p.z, q.z);
  const float hInt  = fminf(p.w, q.w) + fminf(p.y, q.y);
  const float aInt  = wInt * hInt;
  const float aUni  = areaP + areaQ - aInt;
  const float iou   = (aInt + 1.0f) / (aUni + 1.0f);
  unsigned int u = __float_as_uint(iou);
  u = (u & 0x80000000u) ? ~u : (u | 0x80000000u);   // monotone ascending
  keys[i] = ~u;                                     // flip -> descending iou
  vals[i] = __float_as_uint(cen[i]);
}

// ---------------------------------------------------------------------------
// Radix pass: per-block 256-bin histogram (digit-major layout)
// ---------------------------------------------------------------------------
__global__ void ciou_radix_hist(const unsigned int* __restrict__ inKey,
                                unsigned int* __restrict__ hist,
                                int shift, int numBlocks) {
  __shared__ unsigned int cnt[256];
  const int t = threadIdx.x, b = blockIdx.x;
  cnt[t] = 0;
  __syncthreads();
  const unsigned int* tile = inKey + (size_t)b * RDX_TILE;
  for (int i = t; i < RDX_TILE; i += RDX_THREADS)
    atomicAdd(&cnt[(tile[i] >> shift) & 0xFFu], 1u);
  __syncthreads();
  hist[(size_t)t * numBlocks + b] = cnt[t];
}

// ---------------------------------------------------------------------------
// Global exclusive scan of hist[256*numBlocks]  (single workgroup, strips)
// ---------------------------------------------------------------------------
__global__ void ciou_hist_scan(unsigned int* __restrict__ hist, int M) {
  __shared__ unsigned int sums[256];
  const int t = threadIdx.x;
  const int strip = M / 256;
  unsigned int s = 0;
  for (int i = 0; i < strip; ++i) s += hist[(size_t)t * strip + i];
  sums[t] = s;
  __syncthreads();
  if (t == 0) {
    unsigned int run = 0;
    for (int i = 0; i < 256; ++i) { unsigned int x = sums[i]; sums[i] = run; run += x; }
  }
  __syncthreads();
  unsigned int run = sums[t];
  for (int i = 0; i < strip; ++i) {
    unsigned int x = hist[(size_t)t * strip + i];
    hist[(size_t)t * strip + i] = run;
    run += x;
  }
}

// ---------------------------------------------------------------------------
// Radix pass: stable scatter. Tile staged into LDS by async DMA, 8 rounds of
// in-LDS stable binary partition (rank via WMMA block scan of per-thread zero
// counts), then per-digit-run scatter stored asynchronously straight from LDS.
// ---------------------------------------------------------------------------
__global__ void ciou_radix_scatter(const unsigned int* __restrict__ inKey,
                                   const unsigned int* __restrict__ inVal,
                                   unsigned int* __restrict__ outKey,
                                   unsigned int* __restrict__ outVal,
                                   const unsigned int* __restrict__ histScanned,
                                   int shift, int numBlocks) {
  __shared__ unsigned int lkey[RDX_TILE];
  __shared__ unsigned int lval[RDX_TILE];
  __shared__ float sbuf[256];
  __shared__ float sscan[256];
  __shared__ float scol[17];
  __shared__ unsigned int digitStart[256];

  const int t = threadIdx.x, b = blockIdx.x;
  const size_t tileBase = (size_t)b * RDX_TILE;

  const uint4* inK4 = (const uint4*)(inKey + tileBase);
  const uint4* inV4 = (const uint4*)(inVal + tileBase);
  uint4* lk4 = (uint4*)lkey;
  uint4* lv4 = (uint4*)lval;
#ifdef CIOU_ASYNC_LOAD
  // DMA the 32KB tile straight into LDS (no VGPR round-trip); ASYNCcnt tracked.
  for (int i = t; i < RDX_TILE / 4; i += RDX_THREADS) {
    __builtin_amdgcn_global_load_async_to_lds_b128((glb_v4i_ptr)(inK4 + i),
                                                   (lds_v4i_ptr)(lk4 + i), 0, 0);
    __builtin_amdgcn_global_load_async_to_lds_b128((glb_v4i_ptr)(inV4 + i),
                                                   (lds_v4i_ptr)(lv4 + i), 0, 0);
  }
  __builtin_amdgcn_s_wait_asynccnt(0);
  __syncthreads();
#else
  for (int i = t; i < RDX_TILE / 4; i += RDX_THREADS) { lk4[i] = inK4[i]; lv4[i] = inV4[i]; }
  __syncthreads();
#endif

  const int base = t * RDX_ITEMS;
  for (int bit = shift; bit < shift + 8; ++bit) {
    unsigned int k[RDX_ITEMS], v[RDX_ITEMS];
    int zeros = 0;
#pragma unroll
    for (int i = 0; i < RDX_ITEMS; ++i) {
      k[i] = lkey[base + i];
      v[i] = lval[base + i];
      zeros += ((k[i] >> bit) & 1u) ? 0 : 1;
    }
    __syncthreads();   // everyone done reading before in-place scatter

    float total;
    const float incl = blockScan256((float)zeros, sbuf, sscan, scol, &total);
    const int Z = (int)(total + 0.5f);
    int zRun = (int)(incl + 0.5f) - zeros;   // zeros strictly before this thread

#pragma unroll
    for (int i = 0; i < RDX_ITEMS; ++i) {
      const int p = base + i;
      int pos;
      if ((k[i] >> bit) & 1u) { pos = Z + (p - zRun); }
      else                    { pos = zRun++; }
      lkey[pos] = k[i];
      lval[pos] = v[i];
    }
    __syncthreads();
  }

  // tile now digit-sorted & stable; find digit run starts
  digitStart[t] = 0u;
  __syncthreads();
  for (int i = t; i < RDX_TILE; i += RDX_THREADS) {
    const unsigned int d = (lkey[i] >> shift) & 0xFFu;
    if (i == 0 || d != ((lkey[i - 1] >> shift) & 0xFFu)) digitStart[d] = (unsigned int)i;
  }
  __syncthreads();
#ifdef CIOU_ASYNC_STORE
  // Store the sorted runs straight from LDS to their global digit slots.
  for (int i = t; i < RDX_TILE; i += RDX_THREADS) {
    const unsigned int key = lkey[i];
    const unsigned int d   = (key >> shift) & 0xFFu;
    const unsigned int pos = histScanned[(size_t)d * numBlocks + b] + ((unsigned int)i - digitStart[d]);
    __builtin_amdgcn_global_store_async_from_lds_b32((glb_i32_ptr)(outKey + pos),
                                                     (lds_i32_ptr)(lkey + i), 0, 0);
    __builtin_amdgcn_global_store_async_from_lds_b32((glb_i32_ptr)(outVal + pos),
                                                     (lds_i32_ptr)(lval + i), 0, 0);
  }
  __builtin_amdgcn_s_wait_asynccnt(0);
#else
  for (int i = t; i < RDX_TILE; i += RDX_THREADS) {
    const unsigned int key = lkey[i];
    const unsigned int d   = (key >> shift) & 0xFFu;
    const unsigned int pos = histScanned[(size_t)d * numBlocks + b] + ((unsigned int)i - digitStart[d]);
    outKey[pos] = key;
    outVal[pos] = lval[i];
  }
#endif
}

// ---------------------------------------------------------------------------
// K1: per-block double sums of e = exp(-s) over sorted order
// ---------------------------------------------------------------------------
__global__ void ciou_block_esum(const unsigned int* __restrict__ svals,
                                double* __restrict__ blockSum) {
  __shared__ double red[256];
  const int t = threadIdx.x, b = blockIdx.x;
  const float4* tile = (const float4*)((const float*)svals + (size_t)b * RDX_TILE);
  double acc = 0.0;
  for (int i = t; i < RDX_TILE / 4; i += RDX_THREADS) {
    const float4 s4 = tile[i];
    acc += (double)expf(-s4.x) + (double)expf(-s4.y) +
           (double)expf(-s4.z) + (double)expf(-s4.w);
  }
  red[t] = acc;
  __syncthreads();
  for (int s = 128; s > 0; s >>= 1) { if (t < s) red[t] += red[t + s]; __syncthreads(); }
  if (t == 0) blockSum[b] = red[0];
}

// ---------------------------------------------------------------------------
// K2: exclusive scan of block sums (double) + grand total at blockOff[B]
// ---------------------------------------------------------------------------
__global__ void ciou_scan_block_sums(const double* __restrict__ blockSum,
                                     double* __restrict__ blockOff, int B) {
  __shared__ double sums[256];
  const int t = threadIdx.x;
  const int strip = B / 256;
  double s = 0.0;
  for (int i = 0; i < strip; ++i) s += blockSum[t * strip + i];
  sums[t] = s;
  __syncthreads();
  if (t == 0) {
    double run = 0.0;
    for (int i = 0; i < 256; ++i) { double x = sums[i]; sums[i] = run; run += x; }
  }
  __syncthreads();
  double run = sums[t];
  for (int i = 0; i < strip; ++i) {
    double x = blockSum[t * strip + i];
    blockOff[t * strip + i] = run;
    run += x;
  }
  if (t == 255) blockOff[B] = run;   // grand total T
}

// ---------------------------------------------------------------------------
// K3: suffix_i = T - inclusive_prefix_i ; accumulate exp(-3 s_i)*suffix/(n-1-i)
// Uses the WMMA block scan for the 256 per-thread partial sums.
// ---------------------------------------------------------------------------
__global__ void ciou_final_terms(const unsigned int* __restrict__ svals,
                                 const double* __restrict__ blockOff,
                                 double* __restrict__ partial,
                                 int n, int numBlocks) {
  __shared__ float sbuf[256];
  __shared__ float sscan[256];
  __shared__ float scol[17];
  __shared__ double red[256];

  const int t = threadIdx.x, b = blockIdx.x;
  const float* sp = (const float*)svals + (size_t)b * RDX_TILE + t * RDX_ITEMS;
  float sv[RDX_ITEMS], ei[RDX_ITEMS];
  const float4* s4 = (const float4*)sp;
#pragma unroll
  for (int i = 0; i < RDX_ITEMS / 4; ++i) {
    const float4 x = s4[i];
    sv[4 * i + 0] = x.x; sv[4 * i + 1] = x.y; sv[4 * i + 2] = x.z; sv[4 * i + 3] = x.w;
  }
  float run = 0.0f;
#pragma unroll
  for (int i = 0; i < RDX_ITEMS; ++i) { run += expf(-sv[i]); ei[i] = run; }  // local inclusive

  float total;
  const float incl = blockScan256(run, sbuf, sscan, scol, &total);
  const float exclThread = incl - run;

  const double baseOff = blockOff[b];
  const double T = blockOff[numBlocks];
  double acc = 0.0;
#pragma unroll
  for (int i = 0; i < RDX_ITEMS; ++i) {
    const long long gi = (long long)b * RDX_TILE + t * RDX_ITEMS + i;
    if (gi < (long long)n - 1) {
      const double P = baseOff + (double)(exclThread + ei[i]);  // inclusive prefix
      const double suffix = T - P;                              // sum over j > i
      const double cnt = (double)(n - 1 - (int)gi);
      acc += (double)expf(-3.0f * sv[i]) * suffix / cnt;
    }
  }
  red[t] = acc;
  __syncthreads();
  for (int s = 128; s > 0; s >>= 1) { if (t < s) red[t] += red[t + s]; __syncthreads(); }
  if (t == 0) partial[b] = red[0];
}

// ---------------------------------------------------------------------------
// K4: final reduce, divide by (n-1)
// ---------------------------------------------------------------------------
__global__ void ciou_final_reduce(const double* __restrict__ partial,
                                  float* __restrict__ out, int B, int n) {
  __shared__ double red[256];
  const int t = threadIdx.x;
  double acc = 0.0;
  for (int i = t; i < B; i += 256) acc += partial[i];
  red[t] = acc;
  __syncthreads();
  for (int s = 128; s > 0; s >>= 1) { if (t < s) red[t] += red[t + s]; __syncthreads(); }
  if (t == 0) out[0] = (float)(red[0] / (double)(n - 1));
}

// ---------------------------------------------------------------------------
extern "C" void kernel_launch(void* const* d_in, const int* in_sizes, int n_in,
                              void* d_out, int out_size, void* d_ws, size_t ws_size,
                              hipStream_t stream) {
  const int n = in_sizes[0];                  // 4,194,304 (multiple of 4096)
  const int numBlocks = n / RDX_TILE;         // 1024

  const float*  cen  = (const float*)d_in[0];
  // d_in[1] (centerness_targets) unused by the reference output
  const float4* boxA = (const float4*)d_in[3];  // reg_targets_flatten
  const float4* boxB = (const float4*)d_in[2];  // box_regression_flatten

  unsigned char* ws = (unsigned char*)d_ws;
  unsigned int* keysA = (unsigned int*)ws;
  unsigned int* valsA = keysA + n;
  unsigned int* keysB = valsA + n;
  unsigned int* valsB = keysB + n;
  unsigned int* hist  = valsB + n;                              // 256*numBlocks
  double* blockSum = (double*)(hist + (size_t)256 * numBlocks); // numBlocks
  double* blockOff = blockSum + numBlocks;                      // numBlocks + 1
  double* partial  = blockOff + numBlocks + 1;                  // numBlocks

  ciou_build_keys<<<n / 256, 256, 0, stream>>>(cen, boxA, boxB, keysA, valsA, n);

  unsigned int* ik = keysA; unsigned int* iv = valsA;
  unsigned int* ok = keysB; unsigned int* ov = valsB;
  for (int pass = 0; pass < 4; ++pass) {
    const int shift = pass * 8;
    ciou_radix_hist<<<numBlocks, RDX_THREADS, 0, stream>>>(ik, hist, shift, numBlocks);
    ciou_hist_scan<<<1, 256, 0, stream>>>(hist, 256 * numBlocks);
    ciou_radix_scatter<<<numBlocks, RDX_THREADS, 0, stream>>>(ik, iv, ok, ov, hist,
                                                              shift, numBlocks);
    unsigned int* tk = ik; ik = ok; ok = tk;
    unsigned int* tv = iv; iv = ov; ov = tv;
  }
  // 4 passes -> sorted data back in keysA/valsA (== ik/iv)

  ciou_block_esum<<<numBlocks, RDX_THREADS, 0, stream>>>(iv, blockSum);
  ciou_scan_block_sums<<<1, 256, 0, stream>>>(blockSum, blockOff, numBlocks);
  ciou_final_terms<<<numBlocks, RDX_THREADS, 0, stream>>>(iv, blockOff, partial, n, numBlocks);
  ciou_final_reduce<<<1, 256, 0, stream>>>(partial, (float*)d_out, numBlocks, n);
}